// VINNetwork_88699664597680
// MI455X (gfx1250) — compile-verified
//
#include <hip/hip_runtime.h>

typedef float v2f __attribute__((ext_vector_type(2)));
typedef float v8f __attribute__((ext_vector_type(8)));

#define THREADS 256
#define ROWS_PER_BLOCK 256
#define KSTEPS 20

__global__ __launch_bounds__(THREADS) void vin_fused_kernel(
    const float* __restrict__ obs,
    const float* __restrict__ Wphi,
    const float* __restrict__ bphi,
    const float* __restrict__ Wlog,
    const float* __restrict__ blog,
    float* __restrict__ outLogit,
    float* __restrict__ outV,
    int B)
{
    __shared__ float sW[48 * 48];        // W_phi  [out=48][in=48]
    __shared__ float sWl[16 * 48];       // W_logit padded to 16 rows (rows 5..15 zero)
    __shared__ float sBphi[48];
    __shared__ float sBlog[16];
    __shared__ float sPhi[ROWS_PER_BLOCK * 48];  // phi staging (GEMM out -> VIP in)

    const int t = threadIdx.x;
    const int blockRow0 = blockIdx.x * ROWS_PER_BLOCK;

    {   // prefetch the obs row this thread will re-read in the VIP phase
        int pr = blockRow0 + t; if (pr > B - 1) pr = B - 1;
        __builtin_prefetch(obs + (size_t)pr * 48, 0, 1);
    }

    // ---- stage weights/biases in LDS ----
    for (int i = t; i < 48 * 48; i += THREADS) sW[i] = Wphi[i];
    for (int i = t; i < 16 * 48; i += THREADS) {
        int r = i / 48, c = i - r * 48;
        sWl[i] = (r < 5) ? Wlog[r * 48 + c] : 0.0f;
    }
    if (t < 48) sBphi[t] = bphi[t];
    if (t < 16) sBlog[t] = (t < 5) ? blog[t] : 0.0f;
    __syncthreads();

    const int lane = t & 31;
    const int wave = t >> 5;
    const int lm   = lane & 15;   // M (A) / N (B,C,D) position within half-wave
    const int hi   = lane >> 4;   // half-wave select

    // ---- WMMA GEMM: each wave does two 16-row batch tiles ----
    #pragma unroll
    for (int s = 0; s < 2; ++s) {
        const int tileRow = wave * 32 + s * 16;   // local row base (0..255)
        const int gRow0   = blockRow0 + tileRow;  // global row base

        int arowIdx = gRow0 + lm; if (arowIdx > B - 1) arowIdx = B - 1;
        const float* arow = obs + (size_t)arowIdx * 48;

        // A fragments: 16x4 f32, VGPR0 = K(2*hi), VGPR1 = K(2*hi+1)  (ISA 7.12.2)
        v2f a[12];
        #pragma unroll
        for (int kt = 0; kt < 12; ++kt)
            a[kt] = *(const v2f*)(arow + kt * 4 + 2 * hi);

        v8f accP0 = {}; v8f accP1 = {}; v8f accP2 = {}; v8f accL = {};

        #pragma unroll
        for (int kt = 0; kt < 12; ++kt) {
            const int kk = kt * 4 + 2 * hi;
            // B fragments: 4x16 f32; lane = N, lo lanes hold K=0,1, hi lanes K=2,3
            v2f b0, b1, b2, bl;
            b0.x = sW[(     lm) * 48 + kk];  b0.y = sW[(     lm) * 48 + kk + 1];
            b1.x = sW[(16 + lm) * 48 + kk];  b1.y = sW[(16 + lm) * 48 + kk + 1];
            b2.x = sW[(32 + lm) * 48 + kk];  b2.y = sW[(32 + lm) * 48 + kk + 1];
            bl.x = sWl[      lm * 48 + kk];  bl.y = sWl[      lm * 48 + kk + 1];
            accP0 = __builtin_amdgcn_wmma_f32_16x16x4_f32(false, a[kt], false, b0, (short)0, accP0, false, false);
            accP1 = __builtin_amdgcn_wmma_f32_16x16x4_f32(false, a[kt], false, b1, (short)0, accP1, false, false);
            accP2 = __builtin_amdgcn_wmma_f32_16x16x4_f32(false, a[kt], false, b2, (short)0, accP2, false, false);
            accL  = __builtin_amdgcn_wmma_f32_16x16x4_f32(false, a[kt], false, bl, (short)0, accL,  false, false);
        }

        // D layout: lane = N (mod 16), VGPR j holds M = j + 8*hi  -> scatter phi into LDS (+bias)
        #pragma unroll
        for (int j = 0; j < 8; ++j) {
            const int M = j + 8 * hi;
            sPhi[(tileRow + M) * 48 +      lm] = accP0[j] + sBphi[     lm];
            sPhi[(tileRow + M) * 48 + 16 + lm] = accP1[j] + sBphi[16 + lm];
            sPhi[(tileRow + M) * 48 + 32 + lm] = accP2[j] + sBphi[32 + lm];
        }
        // logit: only N=0..4 valid; write straight to global
        if (lm < 5) {
            const float bb = sBlog[lm];
            #pragma unroll
            for (int j = 0; j < 8; ++j) {
                const int M = j + 8 * hi;
                if (gRow0 + M < B)
                    outLogit[(size_t)(gRow0 + M) * 5 + lm] = accL[j] + bb;
            }
        }
    }
    __syncthreads();

    // ---- VIP phase: one sample per thread, everything in registers ----
    const int b = blockRow0 + t;
    float rin[16], rout[16], p[16], v[16];
    const float* prow = &sPhi[t * 48];
    #pragma unroll
    for (int i = 0; i < 16; ++i) {
        rin[i] = prow[i]; rout[i] = prow[16 + i]; p[i] = prow[32 + i]; v[i] = 0.0f;
    }

    for (int k = 0; k < KSTEPS; ++k) {
        float nv[16];
        #pragma unroll
        for (int i = 0; i < 4; ++i) {
            #pragma unroll
            for (int j = 0; j < 4; ++j) {
                const int idx = i * 4 + j;
                const float c = p[idx];
                // zero-padded neighbors: out-of-grid candidate is p*0 + 0 = 0
                float m =        (i > 0) ? (c * v[(idx - 4) & 15] + rin[(idx - 4) & 15]) : 0.0f;
                { float cd = (i < 3) ? (c * v[(idx + 4) & 15] + rin[(idx + 4) & 15]) : 0.0f; m = fmaxf(m, cd); }
                { float cd = (j > 0) ? (c * v[(idx - 1) & 15] + rin[(idx - 1) & 15]) : 0.0f; m = fmaxf(m, cd); }
                { float cd = (j < 3) ? (c * v[(idx + 1) & 15] + rin[(idx + 1) & 15]) : 0.0f; m = fmaxf(m, cd); }
                nv[idx] = fmaxf(v[idx], m - rout[idx]);
            }
        }
        #pragma unroll
        for (int i = 0; i < 16; ++i) v[i] = nv[i];
    }

    if (b < B) {
        // agent position: first nonzero of obs channel 1 (elements 3*i+1), default cell 5
        const float* ob = obs + (size_t)b * 48;
        int flat = 16;
        #pragma unroll
        for (int i = 15; i >= 0; --i) {
            if (ob[3 * i + 1] != 0.0f) flat = i;
        }
        if (flat == 16) flat = 5;
        float vv = 0.0f;
        #pragma unroll
        for (int i = 0; i < 16; ++i) vv = (flat == i) ? v[i] : vv;
        outV[b] = vv;
    }
}

extern "C" void kernel_launch(void* const* d_in, const int* in_sizes, int n_in,
                              void* d_out, int out_size, void* d_ws, size_t ws_size,
                              hipStream_t stream) {
    const float* obs  = (const float*)d_in[0];   // [B,48]
    const float* Wphi = (const float*)d_in[1];   // [48,48]
    const float* bphi = (const float*)d_in[2];   // [48]
    const float* Wlog = (const float*)d_in[3];   // [5,48]
    const float* blog = (const float*)d_in[4];   // [5]
    const int B = in_sizes[0] / 48;

    float* out      = (float*)d_out;
    float* outLogit = out;                        // [B,5] flat
    float* outV     = out + (size_t)B * 5;        // [B]

    const int grid = (B + ROWS_PER_BLOCK - 1) / ROWS_PER_BLOCK;
    vin_fused_kernel<<<grid, THREADS, 0, stream>>>(obs, Wphi, bphi, Wlog, blog,
                                                   outLogit, outV, B);
}